// GNNModel_89094801588808
// MI455X (gfx1250) — compile-verified
//
#include <hip/hip_runtime.h>

// =====================================================================
// 3-layer GAT (PyG GATConv, heads=1) + linear head, fp32 end-to-end.
//
// Pipeline per layer (N=50000, E=800000, D=64; self-loops appended):
//   1. gat_gemm_wmma : h = feat @ W^T           (v_wmma_f32_16x16x4_f32)
//   2. att_dots      : s_i = h_i . a_src,  d_i = h_i . a_dst
//   3. init_node     : zero acc / denom / segment-max state
//   4. edge_logit    : logit_e = leaky(s[src]+d[dst], .2); atomic seg-max
//   5. edge_expsum   : e = exp(logit - max[dst]); atomic seg-sum
//   6. edge_scatter  : acc[dst] += (e/denom[dst]) * h[src]  (16 lanes/edge)
//   7. bias_act      : feat' = leaky(acc + b, .01)
// Head: y_i = feat3_i . Wout + bout
//
// Everything is L2-resident (node state ~38MB + edge data ~10MB << 192MB
// L2), so the edge phases are L2-atomic bound; the GEMM (~0.4 GFLOP/layer)
// runs on the WMMA pipe at full fp32 precision.
// =====================================================================

typedef float v2f __attribute__((ext_vector_type(2)));
typedef float v8f __attribute__((ext_vector_type(8)));

#define D 64
#define WAVES_PER_BLOCK 8   // 256 threads = 8 wave32
#define ROWS_PER_BLOCK (WAVES_PER_BLOCK * 16)

// ---------------- fp32 <-> order-preserving u32 (for atomic seg-max) ----
__device__ __forceinline__ unsigned fenc(float f) {
  unsigned u = __float_as_uint(f);
  return (u & 0x80000000u) ? ~u : (u | 0x80000000u);
}
__device__ __forceinline__ float fdec(unsigned u) {
  return (u & 0x80000000u) ? __uint_as_float(u & 0x7fffffffu)
                           : __uint_as_float(~u);
}

__device__ __forceinline__ void atomic_add_f32(float* p, float v) {
  __hip_atomic_fetch_add(p, v, __ATOMIC_RELAXED, __HIP_MEMORY_SCOPE_AGENT);
}
__device__ __forceinline__ void atomic_max_u32(unsigned* p, unsigned v) {
  __hip_atomic_fetch_max(p, v, __ATOMIC_RELAXED, __HIP_MEMORY_SCOPE_AGENT);
}

// ---------------- 1. h = X @ W^T via V_WMMA_F32_16X16X4_F32 -------------
// Per wave: one 16-row x 64-col tile of H. 4 accumulators (16x16 each),
// K-loop 64 in steps of 4 -> 64 WMMA issues per wave.
// A layout (16x4 f32): lane m = lane&15; vgpr{0,1} hold K = 2*(lane>>4)+{0,1}.
// B layout (4x16 f32): lane n = lane&15; vgpr{0,1} hold K = 2*(lane>>4)+{0,1}.
// C/D (16x16 f32): vgpr r holds row (r + 8*(lane>>4)), col lane&15.
__global__ void gat_gemm_wmma(const float* __restrict__ X,
                              const float* __restrict__ W,
                              float* __restrict__ H, int n) {
  __shared__ float Wlds[D * D];  // 16 KB of the 320 KB WGP LDS pool
  {
    const float4* Wv = (const float4*)W;
    float4* Lv = (float4*)Wlds;
    for (int i = threadIdx.x; i < D * D / 4; i += blockDim.x) Lv[i] = Wv[i];
  }
  __syncthreads();

  const int wave  = threadIdx.x >> 5;
  const int lane  = threadIdx.x & 31;
  const int half  = lane >> 4;   // 0: K pair {0,1} / rows 0-7 ; 1: {2,3} / rows 8-15
  const int l16   = lane & 15;

  const int rowBase = blockIdx.x * ROWS_PER_BLOCK + wave * 16;
  if (rowBase >= n) return;  // wave-uniform exit (after barrier)

  const int m  = rowBase + l16;
  const int ml = (m < n) ? m : (n - 1);  // clamp loads: EXEC stays all-ones
  const float* xrow = X + (size_t)ml * D;

  v8f acc0 = {}, acc1 = {}, acc2 = {}, acc3 = {};

#pragma unroll
  for (int k0 = 0; k0 < D; k0 += 4) {
    const int ka = k0 + 2 * half;
    v2f a;
    a.x = xrow[ka];
    a.y = xrow[ka + 1];

    v2f b0, b1, b2, b3;
    b0.x = Wlds[(0 * 16 + l16) * D + ka];
    b0.y = Wlds[(0 * 16 + l16) * D + ka + 1];
    b1.x = Wlds[(1 * 16 + l16) * D + ka];
    b1.y = Wlds[(1 * 16 + l16) * D + ka + 1];
    b2.x = Wlds[(2 * 16 + l16) * D + ka];
    b2.y = Wlds[(2 * 16 + l16) * D + ka + 1];
    b3.x = Wlds[(3 * 16 + l16) * D + ka];
    b3.y = Wlds[(3 * 16 + l16) * D + ka + 1];

    acc0 = __builtin_amdgcn_wmma_f32_16x16x4_f32(false, a, false, b0,
                                                 (short)0, acc0, false, false);
    acc1 = __builtin_amdgcn_wmma_f32_16x16x4_f32(false, a, false, b1,
                                                 (short)0, acc1, false, false);
    acc2 = __builtin_amdgcn_wmma_f32_16x16x4_f32(false, a, false, b2,
                                                 (short)0, acc2, false, false);
    acc3 = __builtin_amdgcn_wmma_f32_16x16x4_f32(false, a, false, b3,
                                                 (short)0, acc3, false, false);
  }

#pragma unroll
  for (int r = 0; r < 8; ++r) {
    const int row = rowBase + r + 8 * half;
    if (row < n) {
      float* hrow = H + (size_t)row * D;
      hrow[0 * 16 + l16] = acc0[r];
      hrow[1 * 16 + l16] = acc1[r];
      hrow[2 * 16 + l16] = acc2[r];
      hrow[3 * 16 + l16] = acc3[r];
    }
  }
}

// ---------------- 2. per-node attention dot products --------------------
__global__ void att_dots(const float* __restrict__ H,
                         const float* __restrict__ a_src,
                         const float* __restrict__ a_dst,
                         float* __restrict__ S, float* __restrict__ Dv, int n) {
  const int i = blockIdx.x * blockDim.x + threadIdx.x;
  if (i >= n) return;
  const float4* hv = (const float4*)(H + (size_t)i * D);
  const float4* av = (const float4*)a_src;
  const float4* bv = (const float4*)a_dst;
  float ss = 0.f, dd = 0.f;
#pragma unroll
  for (int j = 0; j < D / 4; ++j) {
    float4 h4 = hv[j], a4 = av[j], b4 = bv[j];
    ss += h4.x * a4.x + h4.y * a4.y + h4.z * a4.z + h4.w * a4.w;
    dd += h4.x * b4.x + h4.y * b4.y + h4.z * b4.z + h4.w * b4.w;
  }
  S[i] = ss;
  Dv[i] = dd;
}

// ---------------- 3. zero accumulators / softmax state ------------------
__global__ void init_node(float* __restrict__ acc, unsigned* __restrict__ menc,
                          float* __restrict__ denom, int n) {
  const int i = blockIdx.x * blockDim.x + threadIdx.x;
  if (i >= n) return;
  menc[i] = 0u;       // < fenc(-inf): any logit wins
  denom[i] = 0.f;
  float4 z = {0.f, 0.f, 0.f, 0.f};
  float4* a = (float4*)(acc + (size_t)i * D);
#pragma unroll
  for (int j = 0; j < D / 4; ++j) a[j] = z;
}

// ---------------- 4. edge logits + segment max --------------------------
__global__ void edge_logit(const int* __restrict__ ei,
                           const float* __restrict__ S,
                           const float* __restrict__ Dv,
                           float* __restrict__ elog,
                           unsigned* __restrict__ menc, int E, int n) {
  const int e = blockIdx.x * blockDim.x + threadIdx.x;
  if (e >= E + n) return;
  const int src = (e < E) ? ei[e] : (e - E);       // self-loops appended
  const int dst = (e < E) ? ei[E + e] : (e - E);
  float v = S[src] + Dv[dst];
  v = (v > 0.f) ? v : 0.2f * v;                     // attention leaky_relu
  elog[e] = v;
  atomic_max_u32(menc + dst, fenc(v));
}

// ---------------- 5. exp + segment sum ----------------------------------
__global__ void edge_expsum(const int* __restrict__ ei,
                            const unsigned* __restrict__ menc,
                            float* __restrict__ elog,
                            float* __restrict__ denom, int E, int n) {
  const int e = blockIdx.x * blockDim.x + threadIdx.x;
  if (e >= E + n) return;
  const int dst = (e < E) ? ei[E + e] : (e - E);
  const float ex = expf(elog[e] - fdec(menc[dst]));
  elog[e] = ex;
  atomic_add_f32(denom + dst, ex);
}

// ---------------- 6. weighted scatter: acc[dst] += coef * h[src] --------
// 16 threads per edge, float4 gather per thread -> coalesced 256B/edge.
__global__ void edge_scatter(const int* __restrict__ ei,
                             const float* __restrict__ elog,
                             const float* __restrict__ denom,
                             const float* __restrict__ H,
                             float* __restrict__ acc, int E, int n) {
  const long tid = (long)blockIdx.x * blockDim.x + threadIdx.x;
  if (tid >= (long)(E + n) * 16) return;
  const int e = (int)(tid >> 4);
  const int part = (int)(tid & 15);
  const int src = (e < E) ? ei[e] : (e - E);
  const int dst = (e < E) ? ei[E + e] : (e - E);
  const float coef = elog[e] / denom[dst];
  const float4 h4 = ((const float4*)(H + (size_t)src * D))[part];
  float* o = acc + (size_t)dst * D + part * 4;
  atomic_add_f32(o + 0, coef * h4.x);
  atomic_add_f32(o + 1, coef * h4.y);
  atomic_add_f32(o + 2, coef * h4.z);
  atomic_add_f32(o + 3, coef * h4.w);
}

// ---------------- 7. bias + LeakyReLU(0.01) -----------------------------
__global__ void bias_act(const float* __restrict__ acc,
                         const float* __restrict__ b,
                         float* __restrict__ feat, int n) {
  const int i = blockIdx.x * blockDim.x + threadIdx.x;
  if (i >= n * D) return;
  float v = acc[i] + b[i & (D - 1)];
  feat[i] = (v > 0.f) ? v : 0.01f * v;
}

// ---------------- head: y = feat @ Wout^T + bout ------------------------
__global__ void head(const float* __restrict__ feat,
                     const float* __restrict__ Wout,
                     const float* __restrict__ bout,
                     float* __restrict__ y, int n) {
  const int i = blockIdx.x * blockDim.x + threadIdx.x;
  if (i >= n) return;
  const float4* hv = (const float4*)(feat + (size_t)i * D);
  const float4* wv = (const float4*)Wout;
  float s = 0.f;
#pragma unroll
  for (int j = 0; j < D / 4; ++j) {
    float4 h4 = hv[j], w4 = wv[j];
    s += h4.x * w4.x + h4.y * w4.y + h4.z * w4.z + h4.w * w4.w;
  }
  y[i] = s + bout[0];
}

// =====================================================================
extern "C" void kernel_launch(void* const* d_in, const int* in_sizes, int n_in,
                              void* d_out, int out_size, void* d_ws,
                              size_t ws_size, hipStream_t stream) {
  (void)n_in; (void)out_size; (void)ws_size;

  const float* x = (const float*)d_in[0];
  const float* W[3]  = {(const float*)d_in[1], (const float*)d_in[5],
                        (const float*)d_in[9]};
  const float* As[3] = {(const float*)d_in[2], (const float*)d_in[6],
                        (const float*)d_in[10]};
  const float* Ad[3] = {(const float*)d_in[3], (const float*)d_in[7],
                        (const float*)d_in[11]};
  const float* B[3]  = {(const float*)d_in[4], (const float*)d_in[8],
                        (const float*)d_in[12]};
  const float* Wout = (const float*)d_in[13];
  const float* bout = (const float*)d_in[14];
  const int*   ei   = (const int*)d_in[15];

  const int n = in_sizes[0] / D;     // 50000
  const int E = in_sizes[15] / 2;    // 800000

  // Workspace carve-up (fp32 words): 3*n*D + 4*n + (E+n)  ~= 43 MB.
  float* ws   = (float*)d_ws;
  const size_t nd = (size_t)n * D;
  float*    H    = ws;                       // n*D : transformed features
  float*    ACC  = H + nd;                   // n*D : aggregation accumulator
  float*    FEAT = ACC + nd;                 // n*D : activated layer output
  float*    S    = FEAT + nd;                // n   : h . a_src
  float*    Dv   = S + n;                    // n   : h . a_dst
  unsigned* MENC = (unsigned*)(Dv + n);      // n   : encoded segment max
  float*    DEN  = (float*)(MENC + n);       // n   : softmax denom
  float*    ELOG = DEN + n;                  // E+n : logit, then exp value

  const int tot = E + n;
  const dim3 blk(256);
  const int gGemm  = (n + ROWS_PER_BLOCK - 1) / ROWS_PER_BLOCK;
  const int gNode  = (n + 255) / 256;
  const int gElem  = (n * D + 255) / 256;
  const int gEdge  = (tot + 255) / 256;
  const long scat  = (long)tot * 16;
  const int gScat  = (int)((scat + 255) / 256);

  const float* feat_in = x;
  for (int l = 0; l < 3; ++l) {
    gat_gemm_wmma<<<gGemm, blk, 0, stream>>>(feat_in, W[l], H, n);
    att_dots<<<gNode, blk, 0, stream>>>(H, As[l], Ad[l], S, Dv, n);
    init_node<<<gNode, blk, 0, stream>>>(ACC, MENC, DEN, n);
    edge_logit<<<gEdge, blk, 0, stream>>>(ei, S, Dv, ELOG, MENC, E, n);
    edge_expsum<<<gEdge, blk, 0, stream>>>(ei, MENC, ELOG, DEN, E, n);
    edge_scatter<<<gScat, blk, 0, stream>>>(ei, ELOG, DEN, H, ACC, E, n);
    bias_act<<<gElem, blk, 0, stream>>>(ACC, B[l], FEAT, n);
    feat_in = FEAT;
  }
  head<<<gNode, blk, 0, stream>>>(FEAT, Wout, bout, (float*)d_out, n);
}